// Proto_60292750901307
// MI455X (gfx1250) — compile-verified
//
#include <hip/hip_runtime.h>

typedef __attribute__((ext_vector_type(2))) float v2f;
typedef __attribute__((ext_vector_type(8))) float v8f;

#define B_  32
#define N_  20
#define K_  5
#define Q_  15
#define H_  256
#define M_  (N_ * Q_)          // 300 queries per batch
#define MT_ ((M_ + 15) / 16)   // 19 M-tiles
#define NT_ ((N_ + 15) / 16)   // 2 N-tiles

// ---- Kernel 1: proto[b,n,h] = mean over K of support[b,n,k,h] ----
__global__ __launch_bounds__(256) void proto_mean_kernel(
    const float* __restrict__ support, float* __restrict__ proto) {
  int idx = blockIdx.x * blockDim.x + threadIdx.x;
  if (idx >= B_ * N_ * H_) return;
  int h  = idx & (H_ - 1);
  int bn = idx >> 8;                       // b*N_ + n
  const float* s = support + (size_t)bn * K_ * H_ + h;
  float acc = 0.f;
  #pragma unroll
  for (int k = 0; k < K_; ++k) acc += s[k * H_];
  proto[idx] = acc * (1.0f / K_);
}

// ---- Kernel 2/3: per-row squared L2 norms over H=256 ----
__global__ __launch_bounds__(256) void row_norm_kernel(
    const float* __restrict__ x, float* __restrict__ out, int rows) {
  int r = blockIdx.x * blockDim.x + threadIdx.x;
  if (r >= rows) return;
  const float* p = x + (size_t)r * H_;
  float acc = 0.f;
  #pragma unroll 8
  for (int k = 0; k < H_; ++k) acc += p[k] * p[k];
  out[r] = acc;
}

// ---- Kernel 4: logits via f32 WMMA GEMM + norm correction ----
// One wave32 per 16x16 output tile. D = A(16xK) * B(Kx16) accumulated over
// K=256 in steps of 4 with V_WMMA_F32_16X16X4_F32.
//
// f32 A-matrix 16x4 layout (ISA 7.12.2): lane l holds row m = l&15;
//   VGPR0 = K = (l<16 ? 0 : 2), VGPR1 = K = (l<16 ? 1 : 3).
// f32 B-matrix 4x16 mirrors with lane -> column n = l&15.
// f32 C/D 16x16: VGPR v, lane l -> (M = v + 8*(l>=16), N = l&15).
__global__ __launch_bounds__(32) void proto_logits_wmma(
    const float* __restrict__ query,  // [B, M, H]
    const float* __restrict__ proto,  // [B, N, H]
    const float* __restrict__ qn,     // [B*M]
    const float* __restrict__ pn,     // [B*N]
    float* __restrict__ out) {        // [B, M, N]
  const int lane = threadIdx.x;
  const int li   = lane & 15;
  const int half = lane >> 4;
  const int mt = blockIdx.x, nt = blockIdx.y, b = blockIdx.z;

  const int mrow = mt * 16 + li;   // query row for A loads
  const int ncol = nt * 16 + li;   // proto row for B loads (its column in B)
  // Clamp reads so EXEC stays all-1s (WMMA requirement); stores are guarded.
  const float* qrow = query + ((size_t)b * M_ + (mrow < M_ ? mrow : M_ - 1)) * H_;
  const float* prow = proto + ((size_t)b * N_ + (ncol < N_ ? ncol : N_ - 1)) * H_;
  const int kb = half * 2;         // lane's K-pair within each 4-wide step

  v8f acc = {};
  #pragma unroll
  for (int k = 0; k < H_; k += 4) {
    v2f a  = *(const v2f*)(qrow + k + kb);   // 8B aligned: k+kb is even
    v2f bb = *(const v2f*)(prow + k + kb);
    acc = __builtin_amdgcn_wmma_f32_16x16x4_f32(
        /*neg_a=*/false, a, /*neg_b=*/false, bb,
        /*c_mod=*/(short)0, acc, /*reuse_a=*/false, /*reuse_b=*/false);
  }

  const float pncol = (ncol < N_) ? pn[b * N_ + ncol] : 0.f;
  #pragma unroll
  for (int v = 0; v < 8; ++v) {
    int row = mt * 16 + v + half * 8;
    if (row < M_ && ncol < N_) {
      // -||p - q||^2 = 2*(p.q) - ||q||^2 - ||p||^2
      out[((size_t)b * M_ + row) * N_ + ncol] =
          2.0f * acc[v] - qn[b * M_ + row] - pncol;
    }
  }
}

extern "C" void kernel_launch(void* const* d_in, const int* in_sizes, int n_in,
                              void* d_out, int out_size, void* d_ws, size_t ws_size,
                              hipStream_t stream) {
  const float* support = (const float*)d_in[0];   // [B*N*K, H]
  const float* query   = (const float*)d_in[1];   // [B*N*Q, H]
  float* out = (float*)d_out;                     // [B, M, N]

  // Workspace layout (floats): proto | pn | qn  -> ~680 KB total
  float* ws    = (float*)d_ws;
  float* proto = ws;                      // B*N*H = 163840
  float* pn    = proto + B_ * N_ * H_;    // B*N   = 640
  float* qn    = pn + B_ * N_;            // B*M   = 9600

  const int protoElems = B_ * N_ * H_;
  proto_mean_kernel<<<(protoElems + 255) / 256, 256, 0, stream>>>(support, proto);
  row_norm_kernel<<<(B_ * N_ + 255) / 256, 256, 0, stream>>>(proto, pn, B_ * N_);
  row_norm_kernel<<<(B_ * M_ + 255) / 256, 256, 0, stream>>>(query, qn, B_ * M_);

  dim3 grid(MT_, NT_, B_);
  proto_logits_wmma<<<grid, 32, 0, stream>>>(query, proto, qn, pn, out);
}